// Distanceminimi_Layer_learned_14267881357681
// MI455X (gfx1250) — compile-verified
//
#include <hip/hip_runtime.h>

// D = A*B + C with V_WMMA_F32_16X16X4_F32 (exact f32 math).
typedef __attribute__((ext_vector_type(2))) float v2f;
typedef __attribute__((ext_vector_type(8))) float v8f;

#define DDIM 64
#define ODIM 256
#define ROWS_PER_BLOCK 128   // 8 waves * 16 rows

__launch_bounds__(256)
__global__ void dist_wmma_kernel(const float* __restrict__ x,
                                 const float* __restrict__ omega,
                                 float* __restrict__ out)
{
    __shared__ float ldsP2[ODIM];   // ||omega_o||^2, 1 KB

    const int tid = threadIdx.x;

    // --- p2[o] = ||omega_o||^2 : one prototype per thread (256 threads == O)
    {
        const float* orow = omega + tid * DDIM;
        float s = 0.f;
        #pragma unroll
        for (int d = 0; d < DDIM; d += 4) {
            float4 v = *(const float4*)(orow + d);
            s += v.x * v.x + v.y * v.y + v.z * v.z + v.w * v.w;
        }
        ldsP2[tid] = s;
    }
    __syncthreads();

    const int lane     = tid & 31;
    const int wave     = tid >> 5;
    const int laneMod  = lane & 15;
    const int laneHalf = lane >> 4;   // 0 -> K={0,1} of each K4 chunk, 1 -> K={2,3}

    const long rowBase = (long)blockIdx.x * ROWS_PER_BLOCK + wave * 16;

    // --- A fragments: 16 chunks of K=4 covering D=64 (ISA 32-bit A 16x4 layout:
    // lanes 0-15 hold row=lane K={k0,k0+1}; lanes 16-31 hold row=lane-16 K={k0+2,k0+3})
    const float* xrow = x + (rowBase + laneMod) * DDIM + laneHalf * 2;
    v2f a[16];
    float x2p = 0.f;
    #pragma unroll
    for (int k = 0; k < 16; ++k) {
        a[k] = *(const v2f*)(xrow + k * 4);
        x2p += a[k].x * a[k].x + a[k].y * a[k].y;
    }
    // each row's 64 elements are split across lane L and L^16 -> combine halves
    float x2row = x2p + __shfl_xor(x2p, 16, 32);     // ||x_row||^2, row = laneMod

    // Broadcast into C/D layout: VGPR i holds M=i (lanes 0-15) and M=i+8 (lanes 16-31)
    float x2b[8];
    #pragma unroll
    for (int i = 0; i < 8; ++i)
        x2b[i] = __shfl(x2row, laneHalf ? (i + 8) : i, 32);

    float* outp = out + (rowBase + laneHalf * 8) * (long)ODIM + laneMod;

    // --- 16 column tiles of 16 prototypes; omega stays hot in WGP$/L2 (64 KB)
    for (int ct = 0; ct < 16; ++ct) {
        const int colBase = ct * 16;
        // B 4x16 layout mirrors A: lanes 0-15 K={k0,k0+1}, lanes 16-31 K={k0+2,k0+3},
        // column n = laneMod; B[k][n] = omega[n][k]
        const float* brow = omega + (colBase + laneMod) * DDIM + laneHalf * 2;

        v8f c = {0.f, 0.f, 0.f, 0.f, 0.f, 0.f, 0.f, 0.f};
        #pragma unroll
        for (int k = 0; k < 16; ++k) {
            v2f b = *(const v2f*)(brow + k * 4);
            c = __builtin_amdgcn_wmma_f32_16x16x4_f32(
                    /*neg_a=*/false, a[k], /*neg_b=*/false, b,
                    /*c_mod=*/(short)0, c, /*reuse_a=*/false, /*reuse_b=*/false);
        }

        const float p2n = ldsP2[colBase + laneMod];
        #pragma unroll
        for (int i = 0; i < 8; ++i) {
            // element (m = i + laneHalf*8, n = laneMod) of this 16x16 tile
            outp[(long)i * ODIM + colBase] = x2b[i] + p2n - 2.0f * c[i];
        }
    }
}

extern "C" void kernel_launch(void* const* d_in, const int* in_sizes, int n_in,
                              void* d_out, int out_size, void* d_ws, size_t ws_size,
                              hipStream_t stream)
{
    const float* x     = (const float*)d_in[0];   // [B,H,W,D] f32
    const float* omega = (const float*)d_in[1];   // [O,D]    f32
    float* out = (float*)d_out;                   // [B,H,W,O,1] f32 (flat N x O)

    const long nPix    = (long)in_sizes[0] / DDIM;          // 262144
    const int  nBlocks = (int)(nPix / ROWS_PER_BLOCK);      // 2048

    dist_wmma_kernel<<<nBlocks, 256, 0, stream>>>(x, omega, out);
}